// StructOnlyClassifier_73873437491706
// MI455X (gfx1250) — compile-verified
//
#include <hip/hip_runtime.h>
#include <hip/hip_bf16.h>

typedef __attribute__((ext_vector_type(2))) float v2f;
typedef __attribute__((ext_vector_type(8))) float v8f;

// ---------------------------------------------------------------------------
// Kernel 0: zero the packed per-graph counters (must run every launch; the
// histogram kernel accumulates with atomics).
// ---------------------------------------------------------------------------
__global__ void zero_counts_kernel(unsigned long long* __restrict__ counts, int b) {
    int i = blockIdx.x * blockDim.x + threadIdx.x;
    if (i < b) counts[i] = 0ull;
}

// ---------------------------------------------------------------------------
// Kernel 1: streaming segment histogram.
// batch is sorted, so each thread owns a contiguous 16-node chunk, run-length
// accumulates into one packed u64 (c0 | c1<<21 | c2<<42) and flushes with a
// single 64-bit integer atomic per segment boundary (deterministic).
// ---------------------------------------------------------------------------
__global__ void hist_kernel(const int* __restrict__ batch,
                            const int* __restrict__ ntype,
                            unsigned long long* __restrict__ counts,
                            int n) {
    int chunk = blockIdx.x * blockDim.x + threadIdx.x;
    int base = chunk * 16;
    if (base >= n) return;

    if (base + 16 <= n) {
        // fast path: 4x b128 loads from each array (64B aligned)
        int bb[16], tt[16];
        const int4* b4 = reinterpret_cast<const int4*>(batch + base);
        const int4* t4 = reinterpret_cast<const int4*>(ntype + base);
#pragma unroll
        for (int v = 0; v < 4; ++v) {
            int4 bx = b4[v];
            bb[4 * v + 0] = bx.x; bb[4 * v + 1] = bx.y;
            bb[4 * v + 2] = bx.z; bb[4 * v + 3] = bx.w;
            int4 tx = t4[v];
            tt[4 * v + 0] = tx.x; tt[4 * v + 1] = tx.y;
            tt[4 * v + 2] = tx.z; tt[4 * v + 3] = tx.w;
        }
        int cur = bb[0];
        unsigned long long acc = 0ull;
#pragma unroll
        for (int i = 0; i < 16; ++i) {
            if (bb[i] != cur) {
                atomicAdd(&counts[cur], acc);
                acc = 0ull;
                cur = bb[i];
            }
            acc += 1ull << (21 * tt[i]);
        }
        atomicAdd(&counts[cur], acc);
    } else {
        // tail path (not hit for N = 16M, kept for generality)
        int lim = n - base;
        int cur = batch[base];
        unsigned long long acc = 0ull;
        for (int i = 0; i < lim; ++i) {
            int g = batch[base + i];
            int t = ntype[base + i];
            if (g != cur) {
                atomicAdd(&counts[cur], acc);
                acc = 0ull;
                cur = g;
            }
            acc += 1ull << (21 * t);
        }
        atomicAdd(&counts[cur], acc);
    }
}

// ---------------------------------------------------------------------------
// Kernel 2: MLP 4 -> 32 -> 16 -> 1 on 16-graph tiles via V_WMMA_F32_16X16X4_F32.
// One wave per tile, 8 waves per block.
//
// A operand layout (16x4 f32): lanes 0-15 hold row M=lane, K={0,1}; lanes
// 16-31 hold row M=lane-16, K={2,3}.
// B operand layout (4x16 f32): lanes 0-15 hold col N=lane, K={0,1}; lanes
// 16-31 hold col N=lane-16, K={2,3}.
// C/D layout (16x16 f32): lane 0-15 -> N=lane, VGPR r -> M=r; lane 16-31 ->
// N=lane-16, VGPR r -> M=8+r.
// ---------------------------------------------------------------------------
__global__ void mlp_wmma_kernel(const unsigned long long* __restrict__ counts,
                                const float* __restrict__ W1, const float* __restrict__ B1,
                                const float* __restrict__ W2, const float* __restrict__ B2,
                                const float* __restrict__ W3, const float* __restrict__ B3,
                                float* __restrict__ out, int numGraphs) {
    __shared__ float smem[8][16 * 33];   // per-wave 16x32 tile, stride 33

    const int lane = threadIdx.x & 31;
    const int wave = threadIdx.x >> 5;
    const int tile = blockIdx.x * 8 + wave;
    const int m0 = tile * 16;
    if (m0 >= numGraphs) return;          // uniform per wave -> EXEC stays full

    const int nsel = lane & 15;           // row (for A) / column (for B, C/D)
    const int half = lane >> 4;           // 0: K={0,1}, 1: K={2,3}
    float* sm = smem[wave];

    // ---- feats -> A operand (with clamp to keep EXEC all-ones) ----
    int row = m0 + nsel;
    if (row >= numGraphs) row = numGraphs - 1;
    const unsigned long long p = counts[row];
    const float c0 = (float)(unsigned)(p & 0x1FFFFFull);
    const float c1 = (float)(unsigned)((p >> 21) & 0x1FFFFFull);
    const float c2 = (float)(unsigned)((p >> 42) & 0x1FFFFFull);
    const float f0 = c0 - 54.5f;
    const float f1 = c1;
    const float f2 = c2;
    const float f3 = (c0 + c1 + c2) - 56.5f;

    v2f a1;
    a1[0] = half ? f2 : f0;
    a1[1] = half ? f3 : f1;

    // ---- layer 1: (16x4) @ (4x32) with two WMMAs ----
    const int kb = 2 * half;              // K base this lane supplies: 0 or 2
    v2f b0, b1v;
    b0[0]  = W1[(kb + 0) * 32 + nsel];
    b0[1]  = W1[(kb + 1) * 32 + nsel];
    b1v[0] = W1[(kb + 0) * 32 + nsel + 16];
    b1v[1] = W1[(kb + 1) * 32 + nsel + 16];

    v8f zero = {};
    v8f d0 = __builtin_amdgcn_wmma_f32_16x16x4_f32(false, a1, false, b0,  (short)0, zero, false, false);
    v8f d1 = __builtin_amdgcn_wmma_f32_16x16x4_f32(false, a1, false, b1v, (short)0, zero, false, false);

    const float bia0 = B1[nsel];
    const float bia1 = B1[nsel + 16];
#pragma unroll
    for (int r = 0; r < 8; ++r) {
        const int rr = r + 8 * half;      // C/D row this VGPR holds
        float u0 = d0[r] + bia0; u0 = u0 > 0.f ? u0 : 0.f;
        float u1 = d1[r] + bia1; u1 = u1 > 0.f ? u1 : 0.f;
        sm[rr * 33 + nsel]      = u0;
        sm[rr * 33 + nsel + 16] = u1;
    }
    // LDS is in-order within a wave; each wave owns its region -> no barrier.

    // ---- layer 2: (16x32) @ (32x16), 8 chained K=4 WMMAs ----
    v8f acc = {};
#pragma unroll
    for (int j = 0; j < 8; ++j) {
        const int kk = 4 * j + 2 * half;  // this lane supplies K = kk, kk+1
        v2f aj, bj;
        aj[0] = sm[nsel * 33 + kk];
        aj[1] = sm[nsel * 33 + kk + 1];
        bj[0] = W2[(kk + 0) * 16 + nsel];
        bj[1] = W2[(kk + 1) * 16 + nsel];
        acc = __builtin_amdgcn_wmma_f32_16x16x4_f32(false, aj, false, bj, (short)0, acc, false, false);
    }

    const float bia2 = B2[nsel];
#pragma unroll
    for (int r = 0; r < 8; ++r) {
        float u = acc[r] + bia2; u = u > 0.f ? u : 0.f;
        sm[(r + 8 * half) * 33 + nsel] = u;   // h2 overwrites h1 cols 0..15 (reads done)
    }

    // ---- layer 3: (16x16) @ (16x1) + b3 ----
    if (lane < 16 && (m0 + lane) < numGraphs) {
        float s = B3[0];
#pragma unroll
        for (int c = 0; c < 16; ++c) s += sm[lane * 33 + c] * W3[c];
        out[m0 + lane] = s;
    }
}

// ---------------------------------------------------------------------------
// Host launcher. Inputs (setup_inputs order):
//   0:x(f32,unused) 1:batch(i32) 2:node_type(i32) 3:num_graphs(i32 scalar)
//   4:W1(4x32) 5:b1(32) 6:W2(32x16) 7:b2(16) 8:W3(16) 9:b3(1)
// d_out: (B,) f32, out_size == B.
// ---------------------------------------------------------------------------
extern "C" void kernel_launch(void* const* d_in, const int* in_sizes, int n_in,
                              void* d_out, int out_size, void* d_ws, size_t ws_size,
                              hipStream_t stream) {
    const int*   batch = (const int*)d_in[1];
    const int*   ntype = (const int*)d_in[2];
    const float* W1 = (const float*)d_in[4];
    const float* b1 = (const float*)d_in[5];
    const float* W2 = (const float*)d_in[6];
    const float* b2 = (const float*)d_in[7];
    const float* W3 = (const float*)d_in[8];
    const float* b3 = (const float*)d_in[9];
    float* out = (float*)d_out;

    const int N = in_sizes[1];       // node count
    const int B = out_size;          // graph count (output is (B,1))

    unsigned long long* counts = (unsigned long long*)d_ws;  // B * 8 bytes

    // 0) zero counters (atomics accumulate; harness does not re-poison)
    zero_counts_kernel<<<(B + 255) / 256, 256, 0, stream>>>(counts, B);

    // 1) streaming packed histogram: 16 nodes per thread
    const int nThreads = (N + 15) / 16;
    hist_kernel<<<(nThreads + 255) / 256, 256, 0, stream>>>(batch, ntype, counts, N);

    // 2) WMMA MLP: one wave per 16-graph tile, 8 waves per block
    const int tiles  = (B + 15) / 16;
    const int blocks = (tiles + 7) / 8;
    mlp_wmma_kernel<<<blocks, 256, 0, stream>>>(counts, W1, b1, W2, b2, W3, b3, out, B);
}